// Model_21844203667996
// MI455X (gfx1250) — compile-verified
//
#include <hip/hip_runtime.h>

typedef __attribute__((ext_vector_type(2))) float v2f;
typedef __attribute__((ext_vector_type(8))) float v8f;

#define BSZ   1024
#define NS    512
#define NP    841
#define KP    848      // 841 padded to multiple of 16
#define NT    32
#define LIN2K 1353
#define MN    (BSZ*NS) // 524288

// workspace float offsets (all multiples of 16 floats -> 64B aligned)
#define OFF_E        0          // e[0..3], eb at [4]
#define OFF_MEMTERM  16         // 33*512 (t=1..32 used)
#define OFF_EFP      16912      // 1024*848
#define OFF_WEFP     885264     // 512*848
#define OFF_WACTP    1319440    // 512*512
#define OFF_BASE     1581584    // 1024*512
#define OFF_ACT      2105872    // 1024*512
#define OFF_PREH     2630160    // 32*1024*512

// ---- fold conv+lin1 into a 4-tap stencil; precompute memory-term vectors ----
__global__ __launch_bounds__(512) void prep_kernel(
    const float* __restrict__ conv_w, const float* __restrict__ conv_b,
    const float* __restrict__ lin1_w, const float* __restrict__ lin1_b,
    const float* __restrict__ memory, const float* __restrict__ w_mem,
    float* __restrict__ ws)
{
    int s = threadIdx.x;
    if (s == 0) {
        float e0=0.f, e1=0.f, e2=0.f, e3=0.f, eb=0.f;
        for (int c = 0; c < 64; ++c) {
            float w = lin1_w[c];
            e0 += conv_w[c*4+0]*w; e1 += conv_w[c*4+1]*w;
            e2 += conv_w[c*4+2]*w; e3 += conv_w[c*4+3]*w;
            eb += conv_b[c]*w;
        }
        ws[OFF_E+0]=e0; ws[OFF_E+1]=e1; ws[OFF_E+2]=e2; ws[OFF_E+3]=e3;
        ws[OFF_E+4]=eb + lin1_b[0];
    }
    // memterm[t][s] = sum_{k=0}^{63-t} memory[s][k+t] * wv[k]
    for (int t = 1; t <= NT; ++t) {
        float acc = 0.f;
        for (int k = 0; k + t <= 63; ++k)
            acc += memory[s*64 + k + t] * w_mem[k];
        ws[OFF_MEMTERM + t*NS + s] = acc;
    }
}

// ---- ef[b,p] = relu(stencil(x)) , padded to 848 cols ----
__global__ __launch_bounds__(256) void ef_kernel(
    const float* __restrict__ x, const float* __restrict__ ws, float* __restrict__ efp)
{
    int b = blockIdx.x;
    float e0 = ws[OFF_E+0], e1 = ws[OFF_E+1], e2 = ws[OFF_E+2], e3 = ws[OFF_E+3];
    float eb = ws[OFF_E+4];
    const float* xb = x + (size_t)b*784;
    for (int p = threadIdx.x; p < KP; p += 256) {
        float v = 0.f;
        if (p < NP) {
            int i = p / 29, j = p % 29;
            float acc = eb;
            int r0 = i-1, c0 = j-1;
            if (r0 >= 0) {
                if (c0 >= 0) acc += xb[r0*28 + c0] * e0;
                if (j  < 28) acc += xb[r0*28 + j ] * e1;
            }
            if (i < 28) {
                if (c0 >= 0) acc += xb[i*28 + c0] * e2;
                if (j  < 28) acc += xb[i*28 + j ] * e3;
            }
            v = fmaxf(acc, 0.f);
        }
        efp[(size_t)b*KP + p] = v;
    }
}

// ---- split lin2_w into padded W_ef (512x848) and W_act (512x512), even strides ----
__global__ void pack_kernel(const float* __restrict__ lin2_w,
                            float* __restrict__ wefp, float* __restrict__ wactp)
{
    int idx = blockIdx.x*256 + threadIdx.x;
    const int T1 = 512*KP;
    if (idx < T1) {
        int n = idx / KP, k = idx % KP;
        wefp[idx] = (k < NP) ? lin2_w[n*LIN2K + k] : 0.f;
    } else {
        int j = idx - T1;
        if (j < 512*512) {
            int n = j >> 9, k = j & 511;
            wactp[j] = lin2_w[n*LIN2K + NP + k];
        }
    }
}

__global__ void bcast_kernel(const float* __restrict__ active_state, float* __restrict__ act)
{
    int idx = blockIdx.x*256 + threadIdx.x;
    if (idx < MN) act[idx] = active_state[idx & 511];
}

// ---- f32 WMMA GEMM: Cout = maybe_relu(A(MxK) * B(NxK)^T + Cin + bias) ----
// block = 8 waves; wave tile 32x32 (2x2 of 16x16x4 WMMA); block tile 64x128
__global__ __launch_bounds__(256) void gemm_wmma_kernel(
    const float* __restrict__ A, int lda,
    const float* __restrict__ B, int ldb, int K,
    const float* __restrict__ Cin, const float* __restrict__ bias,
    float* __restrict__ Cout, int doRelu)
{
    const int N = NS;
    int lane = threadIdx.x & 31;
    int wave = threadIdx.x >> 5;
    int wm = blockIdx.y*64  + (wave >> 2)*32;
    int wn = blockIdx.x*128 + (wave & 3)*32;
    int l16 = lane & 15;
    int hi  = lane >> 4;          // lanes 16-31 carry K+2,K+3 per ISA layout

    const float* a0p = A + (size_t)(wm      + l16)*lda + 2*hi;
    const float* a1p = A + (size_t)(wm + 16 + l16)*lda + 2*hi;
    const float* b0p = B + (size_t)(wn      + l16)*ldb + 2*hi;
    const float* b1p = B + (size_t)(wn + 16 + l16)*ldb + 2*hi;

    v8f acc00 = {}, acc01 = {}, acc10 = {}, acc11 = {};
#pragma unroll 4
    for (int k = 0; k < K; k += 4) {
        v2f a0 = *(const v2f*)(a0p + k);
        v2f a1 = *(const v2f*)(a1p + k);
        v2f b0 = *(const v2f*)(b0p + k);
        v2f b1 = *(const v2f*)(b1p + k);
        acc00 = __builtin_amdgcn_wmma_f32_16x16x4_f32(false, a0, false, b0, (short)0, acc00, false, false);
        acc01 = __builtin_amdgcn_wmma_f32_16x16x4_f32(false, a0, false, b1, (short)0, acc01, false, false);
        acc10 = __builtin_amdgcn_wmma_f32_16x16x4_f32(false, a1, false, b0, (short)0, acc10, false, false);
        acc11 = __builtin_amdgcn_wmma_f32_16x16x4_f32(false, a1, false, b1, (short)0, acc11, false, false);
    }

    // epilogue: C/D layout -> lane L, VGPR v holds (M = v + 8*hi, N = l16)
#pragma unroll
    for (int ti = 0; ti < 2; ++ti) {
#pragma unroll
        for (int tj = 0; tj < 2; ++tj) {
            v8f acc = (ti == 0) ? (tj == 0 ? acc00 : acc01)
                                : (tj == 0 ? acc10 : acc11);
            int col = wn + tj*16 + l16;
            float bv = bias ? bias[col] : 0.f;
#pragma unroll
            for (int v = 0; v < 8; ++v) {
                int row = wm + ti*16 + v + hi*8;
                size_t idx = (size_t)row*N + col;
                float val = acc[v] + bv;
                if (Cin) val += Cin[idx];
                if (doRelu) val = fmaxf(val, 0.f);
                Cout[idx] = val;
            }
        }
    }
}

// ---- active_t = memterm_t + b + sum_{h<t} preHist[h] * wv[64-t+h]  (L2-resident) ----
__global__ __launch_bounds__(256) void active_kernel(
    const float* __restrict__ preHist, const float* __restrict__ memterm,
    const float* __restrict__ w_mem, const float* __restrict__ bsc,
    float* __restrict__ act, int t)
{
    int idx = blockIdx.x*256 + threadIdx.x;
    if (idx >= MN) return;
    int s = idx & 511;
    float acc = memterm[t*NS + s] + bsc[0];
    const float* w = w_mem + (64 - t);   // uniform -> scalar loads
    for (int h = 0; h < t; ++h)
        acc += preHist[(size_t)h*MN + idx] * w[h];
    act[idx] = acc;
}

// ---- out = active_32 @ out_w^T + out_b ----
__global__ __launch_bounds__(512) void out_kernel(
    const float* __restrict__ act, const float* __restrict__ out_w,
    const float* __restrict__ out_b, float* __restrict__ out)
{
    __shared__ float row[NS];
    int b = blockIdx.x;
    row[threadIdx.x] = act[(size_t)b*NS + threadIdx.x];
    __syncthreads();
    if (threadIdx.x < 10) {
        int o = threadIdx.x;
        float acc = out_b[o];
        for (int s2 = 0; s2 < NS; ++s2) acc += row[s2] * out_w[o*NS + s2];
        out[b*10 + o] = acc;
    }
}

extern "C" void kernel_launch(void* const* d_in, const int* in_sizes, int n_in,
                              void* d_out, int out_size, void* d_ws, size_t ws_size,
                              hipStream_t stream)
{
    const float* x            = (const float*)d_in[0];
    const float* conv_w       = (const float*)d_in[1];
    const float* conv_b       = (const float*)d_in[2];
    const float* lin1_w       = (const float*)d_in[3];
    const float* lin1_b       = (const float*)d_in[4];
    const float* lin2_w       = (const float*)d_in[5];
    const float* lin2_b       = (const float*)d_in[6];
    const float* out_w        = (const float*)d_in[7];
    const float* out_b        = (const float*)d_in[8];
    const float* active_state = (const float*)d_in[9];
    const float* memory       = (const float*)d_in[10];
    const float* w_mem        = (const float*)d_in[11];
    const float* bsc          = (const float*)d_in[12];

    float* ws      = (float*)d_ws;
    float* efp     = ws + OFF_EFP;
    float* wefp    = ws + OFF_WEFP;
    float* wactp   = ws + OFF_WACTP;
    float* base    = ws + OFF_BASE;
    float* act     = ws + OFF_ACT;
    float* preH    = ws + OFF_PREH;
    float* memterm = ws + OFF_MEMTERM;

    prep_kernel<<<1, 512, 0, stream>>>(conv_w, conv_b, lin1_w, lin1_b, memory, w_mem, ws);
    ef_kernel<<<BSZ, 256, 0, stream>>>(x, ws, efp);
    int packTot = 512*KP + 512*512;
    pack_kernel<<<(packTot + 255)/256, 256, 0, stream>>>(lin2_w, wefp, wactp);
    bcast_kernel<<<(MN + 255)/256, 256, 0, stream>>>(active_state, act);

    dim3 g(NS/128, BSZ/64);  // (4, 16)
    // base = ef @ W_ef^T + lin2_b   (once; K = 848 padded)
    gemm_wmma_kernel<<<g, 256, 0, stream>>>(efp, KP, wefp, KP, KP, nullptr, lin2_b, base, 0);

    for (int t = 1; t <= NT; ++t) {
        // pre_t = relu(base + active_{t-1} @ W_act^T)
        gemm_wmma_kernel<<<g, 256, 0, stream>>>(act, NS, wactp, NS, NS, base, nullptr,
                                                preH + (size_t)(t-1)*MN, 1);
        // active_t from shifted-window dot over pre history
        active_kernel<<<(MN + 255)/256, 256, 0, stream>>>(preH, memterm, w_mem, bsc, act, t);
    }

    out_kernel<<<BSZ, 512, 0, stream>>>(act, out_w, out_b, (float*)d_out);
}